// GRU_RNN_60232621359353
// MI455X (gfx1250) — compile-verified
//
#include <hip/hip_runtime.h>

// GRU RNN: T=256, B=4096, D=128, H=16.
// Phase 1: gx = feat @ W_ih^T + b_ih  (WMMA f32 16x16x4, memory-bound GEMM)
//          feat read NT (stream, don't pollute L2); gx stored with TH_STORE_WB
//          so it stays dirty/resident in the 192MB L2 for phase 2.
// Phase 2: sequential scan, one wave per 16-batch tile, transposed WMMA
//          formulation gh^T = W_hh x h^T so gates stay in D-layout; gx read
//          NT (read-once, evict-first).

typedef float v2f __attribute__((ext_vector_type(2)));
typedef float v4f __attribute__((ext_vector_type(4)));
typedef float v8f __attribute__((ext_vector_type(8)));

#define T_DIM 256
#define B_DIM 4096
#define D_DIM 128
#define H_DIM 16
#define BTILES (B_DIM / 16)      // 256 batch tiles of 16
#define GX_TILE 768              // floats per (t,btile) tile: 16 batch x 48 gates

__device__ __forceinline__ v8f wmma_k4(v2f a, v2f b, v8f c) {
  // D(16x16,f32) = A(16x4,f32) x B(4x16,f32) + C
  return __builtin_amdgcn_wmma_f32_16x16x4_f32(
      /*neg_a=*/false, a, /*neg_b=*/false, b,
      /*c_mod=*/(short)0, c, /*reuse_a=*/false, /*reuse_b=*/false);
}

__device__ __forceinline__ float fsig(float x) {
  float e = __builtin_amdgcn_exp2f(-x * 1.44269504088896f);
  return __builtin_amdgcn_rcpf(1.0f + e);
}
__device__ __forceinline__ float ftanh(float x) {
  float e = __builtin_amdgcn_exp2f(-x * 2.88539008177793f);
  return __builtin_amdgcn_rcpf(1.0f + e) * 2.0f - 1.0f;
}

// Store 128b with TH_STORE_WB: keep line dirty/HT in far cache (L2) so the
// consumer kernel hits L2 instead of HBM.
__device__ __forceinline__ void store_wb_b128(float* p, v4f d) {
  asm volatile("global_store_b128 %0, %1, off th:TH_STORE_WB"
               :
               : "v"(p), "v"(d));
}

// ---------------------------------------------------------------------------
// Phase 1: gx[t,bt] tile (16 batch x 48) = feat tile @ W_ih^T + b_ih, stored
// transposed/swizzled: gx[((t*256+bt)*3+g)*2+half][b=0..15][j=0..7] (8 floats).
// One wave per tile, grid-stride. W_ih (48x128 f32 = 24KB) staged in LDS
// (compiler then promotes the per-lane A chunks into VGPRs across the loop).
// ---------------------------------------------------------------------------
__global__ __launch_bounds__(256) void gru_gx_kernel(
    const float* __restrict__ feat, const float* __restrict__ W_ih,
    const float* __restrict__ b_ih, float* __restrict__ gx) {
  __shared__ float sW[48 * 128];
  const int tid = threadIdx.x;

  // cooperative LDS fill: 6144 floats = 1536 float4, 6 per thread
  {
    const v4f* src = (const v4f*)W_ih;
    v4f* dst = (v4f*)sW;
#pragma unroll
    for (int i = 0; i < 6; ++i) dst[tid + 256 * i] = src[tid + 256 * i];
  }
  __syncthreads();

  const int lane = tid & 31;
  const int wave = tid >> 5;
  const int lp = lane & 15;    // A: row M=j ; B: col N=b ; D: col N=b
  const int half = lane >> 4;  // K sub-select / M-high select

  // C-init = bias broadcast: lane holds j = v + 8*half
  v8f bias[3];
#pragma unroll
  for (int g = 0; g < 3; ++g)
    bias[g] = *(const v8f*)(b_ih + g * 16 + 8 * half);

  const int wavesTotal = gridDim.x * 8;
  const int numTiles = T_DIM * BTILES;  // 65536
  for (int tile = blockIdx.x * 8 + wave; tile < numTiles; tile += wavesTotal) {
    const int t = tile >> 8;
    const int bt = tile & 255;

    // B operand: feat^T chunks. lane lp reads row b=lp, 2 floats at k=4c+2*half
    const float* frow =
        feat + ((size_t)t * B_DIM + (size_t)bt * 16 + lp) * D_DIM + 2 * half;
    v2f bb[32];
#pragma unroll
    for (int c = 0; c < 32; ++c)
      bb[c] = __builtin_nontemporal_load((const v2f*)(frow + 4 * c));  // stream

    v8f acc0 = bias[0], acc1 = bias[1], acc2 = bias[2];
    const float* sWl = sW + lp * 128 + 2 * half;  // gate stride 16*128
#pragma unroll
    for (int c = 0; c < 32; ++c) {
      v2f a0 = *(const v2f*)(sWl + 0 * 2048 + 4 * c);
      v2f a1 = *(const v2f*)(sWl + 1 * 2048 + 4 * c);
      v2f a2 = *(const v2f*)(sWl + 2 * 2048 + 4 * c);
      acc0 = wmma_k4(a0, bb[c], acc0);
      acc1 = wmma_k4(a1, bb[c], acc1);
      acc2 = wmma_k4(a2, bb[c], acc2);
    }

    // D layout: lane holds b=lp, j=v+8*half -> 8 contiguous j per lane.
    // TH_STORE_WB: keep gx resident in L2 for the scan kernel.
    float* o = gx + (size_t)tile * GX_TILE + half * 128 + lp * 8;
    store_wb_b128(o + 0 * 256 + 0, acc0.lo);
    store_wb_b128(o + 0 * 256 + 4, acc0.hi);
    store_wb_b128(o + 1 * 256 + 0, acc1.lo);
    store_wb_b128(o + 1 * 256 + 4, acc1.hi);
    store_wb_b128(o + 2 * 256 + 0, acc2.lo);
    store_wb_b128(o + 2 * 256 + 4, acc2.hi);
  }
}

// ---------------------------------------------------------------------------
// Phase 2: sequential scan. One wave per batch tile (256 waves total).
// h kept in D-layout regs (lane: b=lp, j=v+8*half). W_hh + b_hh in registers.
// ---------------------------------------------------------------------------
__global__ __launch_bounds__(256) void gru_scan_kernel(
    const float* __restrict__ gx, const float* __restrict__ bmask,
    const float* __restrict__ hx0, const float* __restrict__ W_hh,
    const float* __restrict__ b_hh, float* __restrict__ out) {
  const int tid = threadIdx.x;
  const int lane = tid & 31;
  const int wave = tid >> 5;
  const int lp = lane & 15;
  const int half = lane >> 4;
  const int bt = blockIdx.x * 8 + wave;  // 32 blocks x 8 waves = 256 tiles

  // A-operand weights, resident in regs: wa[g][c] = W_hh[g*16+lp][4c+2*half+{0,1}]
  v2f wa[3][4];
#pragma unroll
  for (int g = 0; g < 3; ++g)
#pragma unroll
    for (int c = 0; c < 4; ++c)
      wa[g][c] = *(const v2f*)(W_hh + (g * 16 + lp) * 16 + 4 * c + 2 * half);
  v8f bh[3];
#pragma unroll
  for (int g = 0; g < 3; ++g)
    bh[g] = *(const v8f*)(b_hh + g * 16 + 8 * half);

  // h init from hx0 in D-layout
  v8f h = *(const v8f*)(hx0 + ((size_t)bt * 16 + lp) * H_DIM + 8 * half);

  const size_t tStride = (size_t)BTILES * GX_TILE;  // 196608 floats per step
  const float* gxl = gx + (size_t)bt * GX_TILE + half * 128 + lp * 8;
  const float* bml = bmask + (size_t)bt * 16 + lp;
  float* outl = out + ((size_t)bt * 16 + lp) * H_DIM + 8 * half;

  // software pipeline: preload step 0 (NT: read-once, evict-first from L2)
  v8f gr = __builtin_nontemporal_load((const v8f*)(gxl + 0));
  v8f gz = __builtin_nontemporal_load((const v8f*)(gxl + 256));
  v8f gn = __builtin_nontemporal_load((const v8f*)(gxl + 512));
  float m = bml[0];

  for (int t = 0; t < T_DIM; ++t) {
    if (t + 8 < T_DIM)  // deep prefetch into L2 (global_prefetch_b8)
      __builtin_prefetch(gxl + (size_t)(t + 8) * tStride, 0, 0);

    v8f pr = {}, pz = {}, pn = {};
    float pm = 0.0f;
    if (t + 1 < T_DIM) {  // issue next-step loads before using current data
      const float* p = gxl + (size_t)(t + 1) * tStride;
      pr = __builtin_nontemporal_load((const v8f*)(p + 0));
      pz = __builtin_nontemporal_load((const v8f*)(p + 256));
      pn = __builtin_nontemporal_load((const v8f*)(p + 512));
      pm = bml[(size_t)(t + 1) * B_DIM];
    }

    // begin-of-episode reset
    float keep = 1.0f - m;
#pragma unroll
    for (int v = 0; v < 8; ++v) h[v] *= keep;

    // Build B operand (h^T, K=j, N=b) from D-layout h via half-wave swap.
    float o_[8];
#pragma unroll
    for (int v = 0; v < 8; ++v) o_[v] = __shfl_xor(h[v], 16, 32);
    v2f b0, b1, b2, b3;
    b0.x = half ? o_[2] : h[0];  b0.y = half ? o_[3] : h[1];  // K=0..3
    b1.x = half ? o_[6] : h[4];  b1.y = half ? o_[7] : h[5];  // K=4..7
    b2.x = half ? h[2] : o_[0];  b2.y = half ? h[3] : o_[1];  // K=8..11
    b3.x = half ? h[6] : o_[4];  b3.y = half ? h[7] : o_[5];  // K=12..15

    // gh^T = W_hh x h^T + b_hh : 3 gates x K=16 (4 chained WMMAs each,
    // 3 independent chains interleave in the matrix pipe)
    v8f ar = bh[0], az = bh[1], an = bh[2];
    ar = wmma_k4(wa[0][0], b0, ar); ar = wmma_k4(wa[0][1], b1, ar);
    ar = wmma_k4(wa[0][2], b2, ar); ar = wmma_k4(wa[0][3], b3, ar);
    az = wmma_k4(wa[1][0], b0, az); az = wmma_k4(wa[1][1], b1, az);
    az = wmma_k4(wa[1][2], b2, az); az = wmma_k4(wa[1][3], b3, az);
    an = wmma_k4(wa[2][0], b0, an); an = wmma_k4(wa[2][1], b1, an);
    an = wmma_k4(wa[2][2], b2, an); an = wmma_k4(wa[2][3], b3, an);

    // gates (torch GRUCell order r,z,n) + state update, all in D-layout
#pragma unroll
    for (int v = 0; v < 8; ++v) {
      float r = fsig(gr[v] + ar[v]);
      float z = fsig(gz[v] + az[v]);
      float n = ftanh(gn[v] + r * an[v]);
      h[v] = z * h[v] + (1.0f - z) * n;
    }

    // out[t, b, j..j+7]: 32B contiguous per lane, never re-read -> NT store
    __builtin_nontemporal_store(h, (v8f*)(outl + (size_t)t * (B_DIM * H_DIM)));

    gr = pr; gz = pz; gn = pn; m = pm;
  }
}

// ---------------------------------------------------------------------------
extern "C" void kernel_launch(void* const* d_in, const int* in_sizes, int n_in,
                              void* d_out, int out_size, void* d_ws,
                              size_t ws_size, hipStream_t stream) {
  const float* feat = (const float*)d_in[0];   // [T,B,D]
  const float* bmask = (const float*)d_in[1];  // [T,B,1]
  const float* hx0 = (const float*)d_in[2];    // [B,H]
  const float* W_ih = (const float*)d_in[3];   // [3H,D]
  const float* W_hh = (const float*)d_in[4];   // [3H,H]
  const float* b_ih = (const float*)d_in[5];   // [3H]
  const float* b_hh = (const float*)d_in[6];   // [3H]
  float* out = (float*)d_out;                  // [T,B,H]
  float* gx = (float*)d_ws;                    // 192 MiB staging (swizzled)

  // Phase 1: 1024 blocks x 8 waves = 8192 waves, 8 tiles each (grid-stride)
  gru_gx_kernel<<<1024, 256, 0, stream>>>(feat, W_ih, b_ih, gx);
  // Phase 2: 32 blocks x 8 waves = 256 waves, one batch-tile each
  gru_scan_kernel<<<32, 256, 0, stream>>>(gx, bmask, hx0, W_hh, b_hh, out);
}